// Mult_quat_32401233281621
// MI455X (gfx1250) — compile-verified
//
#include <hip/hip_runtime.h>

// Quaternion Hamilton product: out = q1 * q2, component order (w, x, y, z).
// Shapes: (64, 100000, 4) fp32 -> 6,400,000 quaternions, each one 16B vector.
//
// Memory-bound streaming kernel (0.58 FLOP/byte; roofline ~13us at 23.3 TB/s):
//  - one GLOBAL_LOAD_B128 per input quaternion, one GLOBAL_STORE_B128 per output
//  - CDNA5 TH=NT temporal hints (307 MB stream > 192 MB L2, zero reuse)
//  - 4 quaternions/thread unrolled for MLP
//  - exact-tile kernel with NO bounds checks (nq divides evenly for this shape);
//    guarded tail kernel only launched if a remainder exists.

typedef float v4f __attribute__((ext_vector_type(4)));

#define UNROLL 4
#define THREADS 256
#define TILE (THREADS * UNROLL)

__device__ __forceinline__ v4f hamilton(v4f a, v4f b)
{
    float w1 = a.x, x1 = a.y, y1 = a.z, z1 = a.w;
    float w2 = b.x, x2 = b.y, y2 = b.z, z2 = b.w;
    v4f r;
    r.x = w1 * w2 - x1 * x2 - y1 * y2 - z1 * z2;  // w
    r.y = w1 * x2 + x1 * w2 + y1 * z2 - z1 * y2;  // x
    r.z = w1 * y2 - x1 * z2 + y1 * w2 + z1 * x2;  // y
    r.w = w1 * z2 + x1 * y2 - y1 * x2 + z1 * w2;  // z
    return r;
}

// Full tiles: no bounds checks, straight-line loads -> math -> stores.
__global__ __launch_bounds__(THREADS) void quat_mul_exact(
    const v4f* __restrict__ q1,
    const v4f* __restrict__ q2,
    v4f* __restrict__ out)
{
    int base = blockIdx.x * TILE + threadIdx.x;

    v4f a[UNROLL], b[UNROLL];

#pragma unroll
    for (int u = 0; u < UNROLL; ++u) {
        int i = base + u * THREADS;
        a[u] = __builtin_nontemporal_load(q1 + i);  // global_load_b128 th:NT
        b[u] = __builtin_nontemporal_load(q2 + i);  // global_load_b128 th:NT
    }

#pragma unroll
    for (int u = 0; u < UNROLL; ++u) {
        int i = base + u * THREADS;
        __builtin_nontemporal_store(hamilton(a[u], b[u]), out + i);  // b128 NT
    }
}

// Remainder (< TILE quaternions), guarded. Not launched when nq % TILE == 0.
__global__ __launch_bounds__(THREADS) void quat_mul_tail(
    const v4f* __restrict__ q1,
    const v4f* __restrict__ q2,
    v4f* __restrict__ out,
    int start, int nq)
{
    for (int i = start + threadIdx.x; i < nq; i += THREADS) {
        v4f a = __builtin_nontemporal_load(q1 + i);
        v4f b = __builtin_nontemporal_load(q2 + i);
        __builtin_nontemporal_store(hamilton(a, b), out + i);
    }
}

extern "C" void kernel_launch(void* const* d_in, const int* in_sizes, int n_in,
                              void* d_out, int out_size, void* d_ws, size_t ws_size,
                              hipStream_t stream)
{
    (void)n_in; (void)d_ws; (void)ws_size; (void)in_sizes;

    const v4f* q1 = (const v4f*)d_in[0];
    const v4f* q2 = (const v4f*)d_in[1];
    v4f* out = (v4f*)d_out;

    int nq = out_size / 4;        // 6,400,000 for the reference shape
    int full_blocks = nq / TILE;  // 6250 (exact for this shape)
    int done = full_blocks * TILE;

    if (full_blocks > 0) {
        quat_mul_exact<<<full_blocks, THREADS, 0, stream>>>(q1, q2, out);
    }
    if (done < nq) {
        quat_mul_tail<<<1, THREADS, 0, stream>>>(q1, q2, out, done, nq);
    }
}